// Model_86131274154547
// MI455X (gfx1250) — compile-verified
//
#include <hip/hip_runtime.h>
#include <math.h>
#include <stdint.h>

// ---------------- problem constants (from reference) ----------------
#define NCROPS 10
#define TT     256
#define FF     2048
#define H1DIM  512
#define H2DIM  128
#define MEMN   60
#define KMAXK  17
#define BS     32                       // 2*B
#define NROWS  (BS * NCROPS * TT)       // 81920 rows of 2048 features
#define MTILE  32                       // rows per block (2 WMMA M-tiles)
#define NCHUNK (FF / 64)                // 32 K-chunks of 64

typedef __attribute__((ext_vector_type(2))) float v2f;
typedef __attribute__((ext_vector_type(4))) float v4f;
typedef __attribute__((ext_vector_type(8))) float v8f;

// D = A(16x4 f32) * B(4x16 f32) + C(16x16 f32)
__device__ __forceinline__ v8f wmma_f32(v2f a, v2f b, v8f c) {
  return __builtin_amdgcn_wmma_f32_16x16x4_f32(false, a, false, b,
                                               (short)0, c, false, false);
}

// CDNA5 async global->LDS copy (16B per lane), tracked by ASYNCcnt.
// VDST = VGPR holding the LDS byte address (low 32 bits of the generic ptr).
__device__ __forceinline__ void async_ld16(const float* g, const float* l) {
  uint32_t loff = (uint32_t)(uintptr_t)l;
  asm volatile("global_load_async_to_lds_b128 %0, %1, off"
               :
               : "v"(loff), "v"(g)
               : "memory");
}
__device__ __forceinline__ void wait_async0() {
  asm volatile("s_wait_asynccnt 0" ::: "memory");
}

// ---------------- zero accumulators ----------------
__global__ __launch_bounds__(256) void k_zero(float* __restrict__ featmag,
                                              float* __restrict__ accs,
                                              float* __restrict__ scores_out) {
  int i = blockIdx.x * blockDim.x + threadIdx.x;
  if (i < BS * TT) { featmag[i] = 0.f; scores_out[i] = 0.f; }
  if (i < 8) accs[i] = 0.f;
}

// ---------------- fused MLP: GEMM1+relu -> GEMM2+relu -> GEMM3+sigmoid
// plus per-row feature L2 norm; inputs are read exactly once, A-tiles are
// staged with double-buffered async global->LDS copies.
__global__ __launch_bounds__(256) void k_fused_mlp(
    const float* __restrict__ inp, const float* __restrict__ W1,
    const float* __restrict__ b1, const float* __restrict__ W2,
    const float* __restrict__ b2, const float* __restrict__ W3,
    const float* __restrict__ b3, float* __restrict__ scores_acc,
    float* __restrict__ featmag_acc) {
  __shared__ float lA[2][MTILE][72];    // double-buffered A chunk: 32 x 64 (+8 pad)
  __shared__ float h1s[MTILE][520];     // relu(h1): 32 x 512 (+8 pad)
  __shared__ float h2s[MTILE][136];     // relu(h2): 32 x 128 (+8 pad)
  __shared__ float sqp[MTILE][8];       // per-row sum-of-squares partials

  const int tid  = threadIdx.x;
  const int wave = tid >> 5;
  const int lane = tid & 31;
  const int ln15 = lane & 15;
  const int hi   = (lane >= 16) ? 1 : 0;
  const size_t rowBase = (size_t)blockIdx.x * MTILE;

  // ---- GEMM1: [32 x 2048] @ [2048 x 512]; wave owns 2 M-tiles x 4 N-tiles
  v8f acc[2][4];
  #pragma unroll
  for (int mt = 0; mt < 2; ++mt)
    #pragma unroll
    for (int nt = 0; nt < 4; ++nt)
      #pragma unroll
      for (int r = 0; r < 8; ++r) acc[mt][nt][r] = 0.f;

  // async loader assignment: 8 threads per row, 8 floats each
  const int ldRow = tid >> 3;           // 0..31
  const int ldK   = (tid & 7) * 8;      // 0,8,...,56
  const float* aPtr = inp + (rowBase + (size_t)ldRow) * FF + ldK;

  // prologue: issue chunk 0
  async_ld16(aPtr, &lA[0][ldRow][ldK]);
  async_ld16(aPtr + 4, &lA[0][ldRow][ldK + 4]);

  float sq = 0.f;
  for (int ic = 0; ic < NCHUNK; ++ic) {
    const int cur = ic & 1;
    wait_async0();
    __syncthreads();                     // chunk `cur` visible to all waves
    if (ic + 1 < NCHUNK) {               // prefetch next chunk into other buffer
      const float* g = aPtr + (size_t)(ic + 1) * 64;
      async_ld16(g, &lA[cur ^ 1][ldRow][ldK]);
      async_ld16(g + 4, &lA[cur ^ 1][ldRow][ldK + 4]);
    }
    // read-back for per-row sum of squares (cheap LDS traffic)
    {
      v4f a0 = *(const v4f*)&lA[cur][ldRow][ldK];
      v4f a1 = *(const v4f*)&lA[cur][ldRow][ldK + 4];
      sq += a0.x * a0.x + a0.y * a0.y + a0.z * a0.z + a0.w * a0.w;
      sq += a1.x * a1.x + a1.y * a1.y + a1.z * a1.z + a1.w * a1.w;
    }
    const int kc = ic * 64;
    for (int ks = 0; ks < 64; ks += 4) {
      const int kk = ks + hi * 2;        // A frag: K = kk, kk+1
      v2f afrag0 = *(const v2f*)&lA[cur][ln15][kk];
      v2f afrag1 = *(const v2f*)&lA[cur][16 + ln15][kk];
      const size_t kg0 = (size_t)(kc + kk);
      #pragma unroll
      for (int nt = 0; nt < 4; ++nt) {
        const int col = (wave * 4 + nt) * 16 + ln15;
        v2f bfrag;                       // B reused by both M-tiles
        bfrag.x = W1[kg0 * H1DIM + col];
        bfrag.y = W1[(kg0 + 1) * H1DIM + col];
        acc[0][nt] = wmma_f32(afrag0, bfrag, acc[0][nt]);
        acc[1][nt] = wmma_f32(afrag1, bfrag, acc[1][nt]);
      }
    }
    __syncthreads();                     // all reads of `cur` done before refill
  }

  // per-row ||feat|| -> crop-mean accumulate
  sqp[ldRow][tid & 7] = sq;
  __syncthreads();
  if (tid < MTILE) {
    float s = 0.f;
    #pragma unroll
    for (int j = 0; j < 8; ++j) s += sqp[tid][j];
    size_t r = rowBase + tid;
    int bidx = (int)(r / (NCROPS * TT));
    int tpos = (int)(r & (TT - 1));
    atomicAdd(&featmag_acc[bidx * TT + tpos], sqrtf(s) * (1.0f / NCROPS));
  }

  // bias + relu -> h1 in LDS  (C layout: VGPR r -> M = r + 8*hi, N = ln15)
  #pragma unroll
  for (int mt = 0; mt < 2; ++mt) {
    const int mbase = mt * 16 + hi * 8;
    #pragma unroll
    for (int nt = 0; nt < 4; ++nt) {
      const int col  = (wave * 4 + nt) * 16 + ln15;
      const float bb = b1[col];
      #pragma unroll
      for (int r = 0; r < 8; ++r) {
        float v = acc[mt][nt][r] + bb;
        h1s[mbase + r][col] = v > 0.f ? v : 0.f;
      }
    }
  }
  __syncthreads();

  // ---- GEMM2: [32 x 512] @ [512 x 128]; wave owns N-tile = wave, 2 M-tiles
  v8f acc2[2];
  #pragma unroll
  for (int mt = 0; mt < 2; ++mt)
    #pragma unroll
    for (int r = 0; r < 8; ++r) acc2[mt][r] = 0.f;
  const int col2 = wave * 16 + ln15;
  for (int ks = 0; ks < H1DIM; ks += 4) {
    const int kk = ks + hi * 2;
    v2f afrag0 = *(const v2f*)&h1s[ln15][kk];
    v2f afrag1 = *(const v2f*)&h1s[16 + ln15][kk];
    v2f bfrag;
    bfrag.x = W2[(size_t)kk * H2DIM + col2];
    bfrag.y = W2[(size_t)(kk + 1) * H2DIM + col2];
    acc2[0] = wmma_f32(afrag0, bfrag, acc2[0]);
    acc2[1] = wmma_f32(afrag1, bfrag, acc2[1]);
  }
  {
    const float bb = b2[col2];
    #pragma unroll
    for (int mt = 0; mt < 2; ++mt) {
      const int mbase = mt * 16 + hi * 8;
      #pragma unroll
      for (int r = 0; r < 8; ++r) {
        float v = acc2[mt][r] + bb;
        h2s[mbase + r][col2] = v > 0.f ? v : 0.f;
      }
    }
  }
  __syncthreads();

  // ---- GEMM3 (128->1) + sigmoid + crop-mean scores ----
  if (tid < MTILE) {
    float s = b3[0];
    for (int k = 0; k < H2DIM; ++k) s += h2s[tid][k] * W3[k];
    float sig = 1.f / (1.f + __expf(-s));
    size_t r = rowBase + tid;
    int bidx = (int)(r / (NCROPS * TT));
    int tpos = (int)(r & (TT - 1));
    atomicAdd(&scores_acc[bidx * TT + tpos], sig * (1.0f / NCROPS));
  }
}

// ---------------- per-sample variable-k top-k + mean of top-k scores ----
__global__ __launch_bounds__(32) void k_topk(
    const float* __restrict__ featmag, const float* __restrict__ mask,
    const float* __restrict__ scores, float* __restrict__ out_base,
    int* __restrict__ k_ws, int* __restrict__ idx_ws) {
  __shared__ float mag[TT];
  __shared__ float m2[TT];
  const int s = blockIdx.x;
  const int tid = threadIdx.x;
  for (int t = tid; t < TT; t += 32) {
    float a = 0.f;
    for (int c = 0; c < NCROPS; ++c)
      a += mask[(size_t)(s * NCROPS + c) * TT + t];   // [32,10,1,256]
    float mm = a * (1.0f / NCROPS);
    m2[t] = mm;
    mag[t] = featmag[s * TT + t] * mm;
  }
  __syncthreads();
  if (tid == 0) {
    float msum = 0.f;
    for (int t = 0; t < TT; ++t) msum += m2[t];
    int k = (int)floorf(msum / 16.0f) + 1;
    if (k < 1) k = 1;
    if (k > KMAXK) k = KMAXK;
    k_ws[s] = k;
    float sc = 0.f;
    for (int j = 0; j < KMAXK; ++j) {           // selection in descending order
      float best = -3.4e38f; int bi = 0;
      for (int t = 0; t < TT; ++t)
        if (mag[t] > best) { best = mag[t]; bi = t; }
      mag[bi] = -3.4e38f;
      idx_ws[s * KMAXK + j] = bi;
      if (j < k) sc += scores[s * TT + bi];
    }
    sc /= (float)k;
    if (s < BS / 2) out_base[16 + s] = sc;      // score_nor
    else            out_base[s - 16] = sc;      // score_abn
  }
}

// ---------------- per-(sample,crop) mean of selected snippets + norm ----
__global__ __launch_bounds__(256) void k_fmean(
    const float* __restrict__ inp, const int* __restrict__ k_ws,
    const int* __restrict__ idx_ws, float* __restrict__ x,
    float* __restrict__ out_base) {
  __shared__ float red[256];
  const int j = blockIdx.x;   // row of x: 0..159 normal, 160..319 abnormal
  const int tid = threadIdx.x;
  int sample, crop;
  if (j < 160) { sample = j / NCROPS;              crop = j % NCROPS; }
  else         { sample = 16 + (j - 160) / NCROPS; crop = (j - 160) % NCROPS; }
  const int k = k_ws[sample];
  float fv[8];
  #pragma unroll
  for (int i = 0; i < 8; ++i) fv[i] = 0.f;
  for (int kk = 0; kk < k; ++kk) {
    int t = idx_ws[sample * KMAXK + kk];
    const float* src =
        inp + ((size_t)((sample * NCROPS + crop) * TT + t)) * FF + tid * 8;
    #pragma unroll
    for (int i = 0; i < 8; ++i) fv[i] += src[i];
  }
  const float inv = 1.0f / (float)k;
  float ss = 0.f;
  float* xr = x + (size_t)j * FF + tid * 8;
  #pragma unroll
  for (int i = 0; i < 8; ++i) { float v = fv[i] * inv; xr[i] = v; ss += v * v; }
  red[tid] = ss;
  __syncthreads();
  for (int off = 128; off > 0; off >>= 1) {
    if (tid < off) red[tid] += red[tid + off];
    __syncthreads();
  }
  if (tid == 0) {
    float n = sqrtf(red[0]);
    if (j < 160) out_base[192 + j] = n;          // fs_nor
    else         out_base[32 + (j - 160)] = n;   // fs_abn
  }
}

// ---------------- per-row memory attention: softmax, recon, s/c losses ----
__global__ __launch_bounds__(256) void k_memrow(const float* __restrict__ x,
                                                const float* __restrict__ memB,
                                                float* __restrict__ accs) {
  __shared__ float dotm[MEMN];
  __shared__ float red[256];
  const int j = blockIdx.x;
  const int tid = threadIdx.x;
  if (tid < MEMN) dotm[tid] = 0.f;
  __syncthreads();
  float xs[8];
  const float* xr = x + (size_t)j * FF + tid * 8;
  #pragma unroll
  for (int i = 0; i < 8; ++i) xs[i] = xr[i];
  for (int m = 0; m < MEMN; ++m) {
    const float* mr = memB + (size_t)m * FF + tid * 8;
    float p = 0.f;
    #pragma unroll
    for (int i = 0; i < 8; ++i) p += xs[i] * mr[i];
    atomicAdd(&dotm[m], p);                       // ds_add_f32 reduction
  }
  __syncthreads();
  if (tid == 0) {
    float mx = -3.4e38f;
    for (int m = 0; m < MEMN; ++m) mx = dotm[m] > mx ? dotm[m] : mx;
    float Z = 0.f;
    for (int m = 0; m < MEMN; ++m) { float e = __expf(dotm[m] - mx); dotm[m] = e; Z += e; }
    float invZ = 1.f / Z;
    for (int m = 0; m < MEMN; ++m) dotm[m] *= invZ;
    atomicAdd(&accs[1], (1.0f - invZ) * (1.0f / 320.0f));   // c_loss (max att = 1/Z)
  }
  __syncthreads();
  float ss = 0.f;
  #pragma unroll
  for (int i = 0; i < 8; ++i) {
    float r = 0.f;
    for (int m = 0; m < MEMN; ++m) r += dotm[m] * memB[(size_t)m * FF + tid * 8 + i];
    float d = xs[i] - r;
    ss += d * d;
  }
  red[tid] = ss;
  __syncthreads();
  for (int off = 128; off > 0; off >>= 1) {
    if (tid < off) red[tid] += red[tid + off];
    __syncthreads();
  }
  if (tid == 0) atomicAdd(&accs[0], red[0] * (1.0f / (320.0f * 2048.0f)));
}

// ---------------- memory uniqueness loss + finalize scalars ----------------
__global__ __launch_bounds__(256) void k_uloss(const float* __restrict__ memB,
                                               const float* __restrict__ accs,
                                               float* __restrict__ out_base) {
  __shared__ float nrm[MEMN];
  __shared__ float red[256];
  const int tid = threadIdx.x;
  if (tid < MEMN) {
    const float* mr = memB + (size_t)tid * FF;
    float s = 0.f;
    for (int f = 0; f < FF; ++f) s += mr[f] * mr[f];
    nrm[tid] = sqrtf(s) + 1e-8f;
  }
  __syncthreads();
  float us = 0.f;
  for (int p = tid; p < MEMN * MEMN; p += 256) {
    int i = p / MEMN, jc = p % MEMN;
    const float* a = memB + (size_t)i * FF;
    const float* b = memB + (size_t)jc * FF;
    float d = 0.f;
    for (int f = 0; f < FF; ++f) d += a[f] * b[f];
    float sim = d / (nrm[i] * nrm[jc]);
    float df = sim - (i == jc ? 1.f : 0.f);
    us += df * df;
  }
  red[tid] = us;
  __syncthreads();
  for (int off = 128; off > 0; off >>= 1) {
    if (tid < off) red[tid] += red[tid + off];
    __syncthreads();
  }
  if (tid == 0) {
    out_base[8544] = accs[0];                         // s_loss
    out_base[8545] = accs[1];                         // c_loss
    out_base[8546] = red[0] * (1.0f / (MEMN * MEMN)); // u_loss
  }
}

extern "C" void kernel_launch(void* const* d_in, const int* in_sizes, int n_in,
                              void* d_out, int out_size, void* d_ws,
                              size_t ws_size, hipStream_t stream) {
  const float* inp  = (const float*)d_in[0];
  const float* mask = (const float*)d_in[1];
  const float* W1   = (const float*)d_in[2];
  const float* b1   = (const float*)d_in[3];
  const float* W2   = (const float*)d_in[4];
  const float* b2   = (const float*)d_in[5];
  const float* W3   = (const float*)d_in[6];
  const float* b3   = (const float*)d_in[7];
  const float* memB = (const float*)d_in[8];
  float* out = (float*)d_out;

  // workspace layout (~2.7 MB)
  char* ws = (char*)d_ws;
  float* featmag = (float*)(ws);            // 8192 f
  float* accs    = (float*)(ws + 32768);    // 8 f   [0]=s_loss acc, [1]=c_loss acc
  int*   k_ws    = (int*)  (ws + 32800);    // 32 i
  int*   idx_ws  = (int*)  (ws + 32928);    // 544 i
  float* x       = (float*)(ws + 35328);    // 320*2048 f (concat fmean)

  float* scores_out = out + 352;            // [32,256] crop-mean scores

  k_zero<<<32, 256, 0, stream>>>(featmag, accs, scores_out);
  k_fused_mlp<<<NROWS / MTILE, 256, 0, stream>>>(inp, W1, b1, W2, b2, W3, b3,
                                                 scores_out, featmag);
  k_topk<<<BS, 32, 0, stream>>>(featmag, mask, scores_out, out, k_ws, idx_ws);
  k_fmean<<<320, 256, 0, stream>>>(inp, k_ws, idx_ws, x, out);
  k_memrow<<<320, 256, 0, stream>>>(x, memB, accs);
  k_uloss<<<1, 256, 0, stream>>>(memB, accs, out);
}